// NTN_7499012899173
// MI455X (gfx1250) — compile-verified
//
#include <hip/hip_runtime.h>

typedef __attribute__((ext_vector_type(2))) float v2f;
typedef __attribute__((ext_vector_type(8))) float v8f;
typedef __attribute__((ext_vector_type(4))) unsigned int v4u;
typedef __attribute__((ext_vector_type(8))) int v8i;
typedef __attribute__((ext_vector_type(4))) int v4i;

#define D 128
#define K 16
#define CHUNKS (D / 4)   // 32 WMMA steps of K=4 over the D dimension
#define WPB 2            // waves per block (64 threads)
#define PADROW 132       // 128 + 4 floats pad -> conflict-free ds_load_b64
#define TILE_F (16 * PADROW) // 2112 floats per LDS tile buffer

// LDS (address space 3) typed pointers: keep every shared access in AS3 so the
// backend emits ds_load_* (DScnt only), never flat_load_*.
typedef __attribute__((address_space(3))) float lds_float;
typedef __attribute__((address_space(3))) v2f   lds_v2f;

__device__ __forceinline__ unsigned lds_offset_of(const lds_float* p) {
    return (unsigned)(unsigned long long)p; // AS3 pointers are 32-bit LDS byte offsets
}

// One TDM descriptor: contiguous 8KB x1 tile -> LDS, padding 4 DWORDs every
// 128 DWORDs so each 128-float row lands at stride 132 (bank-conflict-free).
// D# layout per CDNA5 ISA.
__device__ __forceinline__ void tdm_load_tile(const float* gsrc, unsigned lds_off,
                                              unsigned valid_elems) {
    unsigned long long ga = (unsigned long long)gsrc;
    v4u g0 = { 1u,                                   // count=1 (valid), user mode
               lds_off,                              // lds_addr (bytes)
               (unsigned)ga,                         // global_addr[31:0]
               ((unsigned)(ga >> 32) & 0x1FFFFFFu)   // global_addr[56:32]
                   | (2u << 30) };                   // type=2 ("image")
    v8i g1 = { (int)((2u << 16)     // data_size = 4 bytes
                   | (1u << 20)     // pad_enable
                   | (6u << 22)     // pad_interval: every 128 DWORDs
                   | (3u << 25)),   // pad_amount: 4 DWORDs
               (int)((valid_elems & 0xFFFFu) << 16), // tensor_dim0[15:0] (OOB reads -> 0)
               (int)((valid_elems >> 16) | (1u << 16)), // tensor_dim0[31:16], tensor_dim1=1
               (int)(2048u << 16),  // tile_dim0 = 2048 elements (8KB)
               0,                   // tile_dim1=0 (unused), tile_dim2=0
               2048,                // tensor_dim0_stride (unused for 1 row)
               0, 0 };
    v4i gz4 = { 0, 0, 0, 0 };          // groups 2/3: dims 2..4 unused
    v8i gz8 = { 0, 0, 0, 0, 0, 0, 0, 0 };
    __builtin_amdgcn_tensor_load_to_lds(g0, g1, gz4, gz4, gz8, 0);
}

// ---------------------------------------------------------------------------
// Kernel 1 (tiny): Bcomb[k,d] = V[k,d] + sum_e W[k,d,e]*x2[e]
//                  cvec[k]    = sum_e V[k,D+e]*x2[e] + b[k]
// ---------------------------------------------------------------------------
__global__ void ntn_prep(const float* __restrict__ x2,
                         const float* __restrict__ V,
                         const float* __restrict__ W,
                         const float* __restrict__ b,
                         float* __restrict__ Bcomb, // K*D floats
                         float* __restrict__ cvec)  // K floats
{
    int idx = blockIdx.x * blockDim.x + threadIdx.x; // 0..2047
    if (idx < K * D) {
        int k = idx / D, d = idx % D;
        const float* Wrow = W + (size_t)k * D * D + (size_t)d * D;
        float s = 0.f;
        #pragma unroll 8
        for (int e = 0; e < D; ++e) s += Wrow[e] * x2[e];
        Bcomb[idx] = V[k * (2 * D) + d] + s;
    }
    if (idx < K) {
        const float* Vrow = V + idx * (2 * D) + D;
        float s = 0.f;
        #pragma unroll 8
        for (int e = 0; e < D; ++e) s += Vrow[e] * x2[e];
        cvec[idx] = s + b[idx];
    }
}

// ---------------------------------------------------------------------------
// Kernel 2: one wave32 per 16-row tile (grid-stride). Per tile:
//   TDM async-load of the contiguous 8KB tile into padded LDS (double-buffered,
//   prefetching the next tile), then 32x v_wmma_f32_16x16x4_f32 fed by
//   conflict-free ds_load_b64, then fused relu/bias/U-dot shuffle epilogue.
// ---------------------------------------------------------------------------
__global__ void __launch_bounds__(WPB * 32) ntn_main(
    const float* __restrict__ x1,
    const float* __restrict__ Bcomb,
    const float* __restrict__ cvec,
    const float* __restrict__ U,
    float* __restrict__ out,
    int N, int ntiles)
{
    __shared__ float smem[WPB * 2 * TILE_F]; // 33,792 B

    const int lane = threadIdx.x & 31;
    const int half = lane >> 4;  // K-half within WMMA fragments
    const int l16  = lane & 15;  // A: row M; B/D: column N (= output k)
    const int wave = __builtin_amdgcn_readfirstlane((int)(threadIdx.x >> 5)); // scalar
    const int gwave = blockIdx.x * WPB + wave;
    const int nwaves = (int)gridDim.x * WPB;

    // Stay in address space 3 for all tile-buffer accesses.
    lds_float* wbuf = ((lds_float*)smem) + wave * 2 * TILE_F; // this wave's 2 buffers
    const unsigned ldsoff0 = lds_offset_of(wbuf);
    const unsigned ldsoff1 = ldsoff0 + TILE_F * 4u;

    // Preload all B fragments once: lane L, vgpr j -> Bcomb[l16*D + 4c + j + 2*half]
    v2f bfrag[CHUNKS];
    {
        const v2f* bp = (const v2f*)(Bcomb + l16 * D + 2 * half); // 8B aligned
        #pragma unroll
        for (int c = 0; c < CHUNKS; ++c) bfrag[c] = bp[2 * c];
    }
    const float cv = cvec[l16];
    const float uk = U[l16];

    // Prologue: kick off the first tile's DMA.
    if (gwave < ntiles) {
        unsigned ve = (unsigned)((N - gwave * 16) * D); if (ve > 2048u) ve = 2048u;
        tdm_load_tile(x1 + (size_t)gwave * 16 * D, ldsoff0, ve);
    }

    int cur = 0;
    for (int t = gwave; t < ntiles; t += nwaves) { // wave-uniform: EXEC stays full
        // Prefetch next tile into the other buffer, then release current buffer.
        const int tn = t + nwaves;
        if (tn < ntiles) {
            unsigned ve = (unsigned)((N - tn * 16) * D); if (ve > 2048u) ve = 2048u;
            tdm_load_tile(x1 + (size_t)tn * 16 * D, (cur ? ldsoff0 : ldsoff1), ve);
            __builtin_amdgcn_s_wait_tensorcnt(1); // oldest (current buffer) done
        } else {
            __builtin_amdgcn_s_wait_tensorcnt(0);
        }

        const lds_float* sp = wbuf + cur * TILE_F + l16 * PADROW + 2 * half;
        v8f acc = {0.f, 0.f, 0.f, 0.f, 0.f, 0.f, 0.f, 0.f};
        #pragma unroll
        for (int c = 0; c < CHUNKS; ++c) {
            v2f a = *(const lds_v2f*)(sp + 4 * c); // ds_load_b64, conflict-free (stride 132)
            acc = __builtin_amdgcn_wmma_f32_16x16x4_f32(
                false, a, false, bfrag[c], (short)0, acc, false, false);
        }

        // D layout: acc[r] = dot(x1[16t + r + 8*half], Bcomb[l16])
        const int n0 = t * 16;
        #pragma unroll
        for (int r = 0; r < 8; ++r) {
            float v = acc[r] + cv;
            v = v > 0.f ? v : 0.f;
            v *= uk;
            v += __shfl_xor(v, 1, 32); // reduce 16 k-values within half-wave
            v += __shfl_xor(v, 2, 32);
            v += __shfl_xor(v, 4, 32);
            v += __shfl_xor(v, 8, 32);
            const int n = n0 + r + 8 * half;
            if (l16 == 0 && n < N) out[n] = v;
        }
        cur ^= 1;
    }
}

// ---------------------------------------------------------------------------
extern "C" void kernel_launch(void* const* d_in, const int* in_sizes, int n_in,
                              void* d_out, int out_size, void* d_ws, size_t ws_size,
                              hipStream_t stream) {
    const float* x1 = (const float*)d_in[0]; // (N, D)
    const float* x2 = (const float*)d_in[1]; // (1, D)
    const float* V  = (const float*)d_in[2]; // (K, 2D)
    const float* W  = (const float*)d_in[3]; // (K, D, D)
    const float* b  = (const float*)d_in[4]; // (K,)
    const float* U  = (const float*)d_in[5]; // (K, 1)
    float* out = (float*)d_out;              // (N, 1)

    const int N = in_sizes[0] / D;

    float* Bcomb = (float*)d_ws;   // K*D floats
    float* cvec  = Bcomb + K * D;  // K floats

    ntn_prep<<<8, 256, 0, stream>>>(x2, V, W, b, Bcomb, cvec);

    const int ntiles = (N + 15) / 16; // 31250 for N=500000
    int blocks = (ntiles + WPB - 1) / WPB;
    if (blocks > 8192) blocks = 8192; // grid-stride beyond this
    if (blocks < 1) blocks = 1;
    ntn_main<<<blocks, WPB * 32, 0, stream>>>(x1, Bcomb, cvec, U, out, N, ntiles);
}